// CRZRing_18811956756653
// MI455X (gfx1250) — compile-verified
//
#include <hip/hip_runtime.h>
#include <math.h>

// Problem constants (match the reference).
#define N_WIRES 14
#define DIM     16384
#define BATCH   2048

// Tiling.
#define TILE_D          2048   // d-values per block; cos/sin tile = 16 KB LDS
#define THREADS         256    // 8 waves (wave32)
#define ROWS_PER_BLOCK  8      // batch rows per block
#define CHUNKS          2      // TILE_D / (THREADS * 4)

// Native clang vector types (required by __builtin_nontemporal_*).
typedef float f32x4 __attribute__((ext_vector_type(4)));
typedef float f32x2 __attribute__((ext_vector_type(2)));

// ---------------------------------------------------------------------------
// CDNA5 async global -> LDS copy of 16 bytes per lane (gfx1250 ASYNCcnt path).
// VDST vgpr = workgroup-relative LDS byte address, VADDR = 64-bit global addr.
// ---------------------------------------------------------------------------
__device__ __forceinline__ void async_copy_b128(unsigned lds_byte_off,
                                                const void* gsrc) {
    unsigned long long ga = (unsigned long long)gsrc;
    asm volatile("global_load_async_to_lds_b128 %0, %1, off"
                 :: "v"(lds_byte_off), "v"(ga)
                 : "memory");
}

__device__ __forceinline__ void wait_async_zero() {
    asm volatile("s_wait_asynccnt 0" ::: "memory");
}

// ---------------------------------------------------------------------------
// Complex rotation of two elements packed into one 128-bit store.
// out = (r*c - i*s, r*s + i*c) for each of the two (d, d+1) elements.
// ---------------------------------------------------------------------------
__device__ __forceinline__ f32x4 rot_pair(float r0, float i0, float c0, float s0,
                                          float r1, float i1, float c1, float s1) {
    f32x4 o;
    o.x = fmaf(r0, c0, -(i0 * s0));
    o.y = fmaf(r0, s0,  (i0 * c0));
    o.z = fmaf(r1, c1, -(i1 * s1));
    o.w = fmaf(r1, s1,  (i1 * c1));
    return o;
}

// ---------------------------------------------------------------------------
// Shared streaming body: cs[] is the per-tile {cos,sin} table in LDS.
// Chunk c, lane t handles d_local = c*1024 + t*4 (4 complex elements).
// All global accesses are coalesced 128-bit, marked non-temporal (streamed).
// ---------------------------------------------------------------------------
__device__ __forceinline__ void apply_tile(const f32x2* __restrict__ cs,
                                           const f32x4* __restrict__ real4,
                                           const f32x4* __restrict__ imag4,
                                           f32x4* __restrict__ out4,
                                           int tile0, int tid, int row0) {
    const f32x4* csv = (const f32x4*)cs;   // one f32x4 = 2 {c,s} pairs

    // Hoist this lane's 8 {c,s} pairs into registers (reused over 8 rows).
    f32x4 csA[CHUNKS], csB[CHUNKS];
#pragma unroll
    for (int c = 0; c < CHUNKS; ++c) {
        csA[c] = csv[c * 512 + tid * 2];       // pairs for d+0, d+1
        csB[c] = csv[c * 512 + tid * 2 + 1];   // pairs for d+2, d+3
    }

#pragma unroll
    for (int r = 0; r < ROWS_PER_BLOCK; ++r) {
        const size_t row      = (size_t)row0 + r;
        const size_t in_base  = row * (DIM / 4) + (size_t)(tile0 / 4);
        const size_t out_base = row * (DIM / 2) + (size_t)(tile0 / 2);
#pragma unroll
        for (int c = 0; c < CHUNKS; ++c) {
            const size_t ii = in_base + (size_t)(c * 256 + tid);
            f32x4 rv = __builtin_nontemporal_load(&real4[ii]);
            f32x4 iv = __builtin_nontemporal_load(&imag4[ii]);

            f32x4 o0 = rot_pair(rv.x, iv.x, csA[c].x, csA[c].y,
                                rv.y, iv.y, csA[c].z, csA[c].w);
            f32x4 o1 = rot_pair(rv.z, iv.z, csB[c].x, csB[c].y,
                                rv.w, iv.w, csB[c].z, csB[c].w);

            const size_t oo = out_base + (size_t)(c * 512 + tid * 2);
            __builtin_nontemporal_store(o0, &out4[oo]);
            __builtin_nontemporal_store(o1, &out4[oo + 1]);
        }
    }
}

// ---------------------------------------------------------------------------
// Kernel 1: build the {cos(p), sin(p)} table for all 16384 basis states.
// p[d] = 0.5 * sum_i theta_i * bit_i(d) * (2*bit_{(i+1)%n}(d) - 1), wire 0 = MSB.
// ---------------------------------------------------------------------------
__global__ void __launch_bounds__(256)
crz_phase_table_kernel(const float* __restrict__ theta, f32x2* __restrict__ tab) {
    const int d = blockIdx.x * blockDim.x + threadIdx.x;
    float p = 0.0f;
#pragma unroll
    for (int i = 0; i < N_WIRES; ++i) {
        const int j   = (i + 1) % N_WIRES;
        const int ctl = (d >> (N_WIRES - 1 - i)) & 1;
        const int tgt = (d >> (N_WIRES - 1 - j)) & 1;
        p += theta[i] * (float)(ctl * (2 * tgt - 1));
    }
    p *= 0.5f;
    float s, c;
    __sincosf(p, &s, &c);
    f32x2 v;
    v.x = c;
    v.y = s;
    tab[d] = v;
}

// ---------------------------------------------------------------------------
// Kernel 2 (main path): async-DMA the tile's cos/sin into LDS, then stream.
// ---------------------------------------------------------------------------
__global__ void __launch_bounds__(THREADS)
crz_apply_kernel(const f32x4* __restrict__ real4,
                 const f32x4* __restrict__ imag4,
                 const f32x2* __restrict__ tab,
                 f32x4* __restrict__ out4) {
    __shared__ __align__(16) f32x2 cs[TILE_D];

    const int tid   = threadIdx.x;
    const int tile0 = blockIdx.x * TILE_D;
    const int row0  = blockIdx.y * ROWS_PER_BLOCK;

    // Stage 16 KB of {cos,sin} via CDNA5 async global->LDS (4 x b128 per lane).
    {
        const char*    gsrc     = (const char*)(tab + tile0);
        const unsigned lds_base = (unsigned)(unsigned long long)(const void*)cs;
#pragma unroll
        for (int k = 0; k < (TILE_D * 8) / (THREADS * 16); ++k) {
            const unsigned byte_off = (unsigned)(tid + k * THREADS) * 16u;
            async_copy_b128(lds_base + byte_off, gsrc + byte_off);
        }
        wait_async_zero();
        __syncthreads();
    }

    apply_tile(cs, real4, imag4, out4, tile0, tid, row0);
}

// ---------------------------------------------------------------------------
// Fallback fused kernel (no workspace needed): each block computes its own
// tile's phases into LDS, then runs the same streaming body.
// ---------------------------------------------------------------------------
__global__ void __launch_bounds__(THREADS)
crz_fused_kernel(const f32x4* __restrict__ real4,
                 const f32x4* __restrict__ imag4,
                 const float* __restrict__ theta,
                 f32x4* __restrict__ out4) {
    __shared__ __align__(16) f32x2 cs[TILE_D];

    const int tid   = threadIdx.x;
    const int tile0 = blockIdx.x * TILE_D;
    const int row0  = blockIdx.y * ROWS_PER_BLOCK;

#pragma unroll
    for (int k = 0; k < TILE_D / THREADS; ++k) {
        const int dl = k * THREADS + tid;
        const int d  = tile0 + dl;
        float p = 0.0f;
#pragma unroll
        for (int i = 0; i < N_WIRES; ++i) {
            const int j   = (i + 1) % N_WIRES;
            const int ctl = (d >> (N_WIRES - 1 - i)) & 1;
            const int tgt = (d >> (N_WIRES - 1 - j)) & 1;
            p += theta[i] * (float)(ctl * (2 * tgt - 1));
        }
        p *= 0.5f;
        float s, c;
        __sincosf(p, &s, &c);
        f32x2 v;
        v.x = c;
        v.y = s;
        cs[dl] = v;
    }
    __syncthreads();

    apply_tile(cs, real4, imag4, out4, tile0, tid, row0);
}

// ---------------------------------------------------------------------------
// Host-side launcher.
// ---------------------------------------------------------------------------
extern "C" void kernel_launch(void* const* d_in, const int* in_sizes, int n_in,
                              void* d_out, int out_size, void* d_ws, size_t ws_size,
                              hipStream_t stream) {
    const f32x4* real4 = (const f32x4*)d_in[0];
    const f32x4* imag4 = (const f32x4*)d_in[1];
    const float* theta = (const float*)d_in[2];
    f32x4*       out4  = (f32x4*)d_out;

    const dim3 grid(DIM / TILE_D, BATCH / ROWS_PER_BLOCK);  // (8, 256)

    if (d_ws != nullptr && ws_size >= (size_t)DIM * sizeof(f32x2)) {
        f32x2* tab = (f32x2*)d_ws;
        crz_phase_table_kernel<<<DIM / 256, 256, 0, stream>>>(theta, tab);
        crz_apply_kernel<<<grid, THREADS, 0, stream>>>(real4, imag4, tab, out4);
    } else {
        crz_fused_kernel<<<grid, THREADS, 0, stream>>>(real4, imag4, theta, out4);
    }
}